// PRMS_72670846648962
// MI455X (gfx1250) — compile-verified
//
#include <hip/hip_runtime.h>
#include <math.h>

#define NEARZERO 1e-05f

// Per-element PRMS recurrence. One thread per (grid, mul) element.
// All parameters + 7 state variables live in VGPRs for the whole 365-step scan.
template <int NMUL>
__global__ __launch_bounds__(256) void prms_scan_kernel(
    const float* __restrict__ x,      // [Nstep, Ngrid, 3]  (prcp, tmean, pet)
    const float* __restrict__ plast,  // [Ngrid, 18, nmul]  == params_raw[-1]
    float* __restrict__ out,          // [Nstep, Ngrid, nmul]
    int Ngrid, int nmul_rt, int Nstep)
{
    const int nmul  = (NMUL > 0) ? NMUL : nmul_rt;
    const int total = Ngrid * nmul;
    const int tid   = blockIdx.x * blockDim.x + threadIdx.x;
    if (tid >= total) return;

    const int g = tid / nmul;
    const int m = tid - g * nmul;

    // PARAM_BOUNDS (lo, hi) in order:
    // tt ddf alpha beta stor retip fscn scx flz stot cgw resmax k1 k2 k3 k4 k5 k6
    const float lo[18] = {-3.f, 0.f, 0.f, 0.f, 0.f, 0.f,  0.f, 0.f, 0.005f,    1.f, 0.f,   1.f, 0.f, 1.f, 0.f, 0.f, 0.f, 0.f};
    const float hi[18] = { 5.f,20.f, 1.f, 1.f, 5.f,50.f,  1.f, 1.f, 0.995f, 2000.f,20.f, 300.f, 1.f, 5.f, 1.f, 1.f, 1.f, 1.f};

    float p[18];
    {
        const float* __restrict__ pr = plast + (size_t)g * 18 * nmul + m;
        #pragma unroll
        for (int i = 0; i < 18; ++i)
            p[i] = fmaf(hi[i] - lo[i], pr[(size_t)i * nmul], lo[i]);
    }
    const float tt     = p[0],  ddf   = p[1],  alpha = p[2],  beta  = p[3];
    const float stor   = p[4],  retip = p[5],  fscn  = p[6],  scx   = p[7];
    const float flz    = p[8],  stot  = p[9],  cgw   = p[10], resmax= p[11];
    const float k1     = p[12], k2    = p[13], k3    = p[14], k4    = p[15];
    const float k5     = p[16], k6    = p[17];

    // Time-invariant derived values: hoist every divide out of the scan loop.
    const float scn        = fscn * scx;
    const float scx_m_scn  = scx - scn;
    const float remx       = (1.0f - flz) * stot;
    const float smax       = flz * stot;
    const float inv_remx   = 1.0f / remx;
    const float inv_smax   = 1.0f / smax;
    const float inv_resmax = 1.0f / resmax;
    const float omb        = 1.0f - beta;   // (1 - beta)
    const float oma        = 1.0f - alpha;  // (1 - alpha)

    // State (7 registers)
    float snow = 0.001f, XIN = 0.001f, RSTOR = 0.001f, RECHR = 0.001f;
    float SMAV = 0.001f, RES = 0.001f, GW = 0.001f;

    const float* __restrict__ xg = x + (size_t)g * 3;
    const size_t xstride = (size_t)Ngrid * 3;

    // Software-pipelined forcing load (register prefetch of t+1).
    float P = xg[0], T = xg[1], E = xg[2];

    for (int t = 0; t < Nstep; ++t) {
        // Prefetch next step into registers; deeper prefetch into cache
        // (each wave is ~alone on its SIMD, so latency must be hidden in-wave).
        const int tn = (t + 1 < Nstep) ? (t + 1) : t;
        const float* __restrict__ xn = xg + (size_t)tn * xstride;
        const float Pn = xn[0], Tn = xn[1], En = xn[2];
        if (t + 6 < Nstep)
            __builtin_prefetch(xg + (size_t)(t + 6) * xstride, 0, 1);  // global_prefetch_b8

        // ---- snow partition & melt ----
        const bool  cold    = (T <= tt);
        const float flux_ps = cold ? P : 0.0f;
        const float flux_pr = cold ? 0.0f : P;
        snow += flux_ps;
        float flux_m = fmaxf(ddf * (T - tt), 0.0f);
        flux_m = fminf(flux_m, snow);
        snow   = fmaxf(snow - flux_m, NEARZERO);

        // ---- canopy ----
        const float flux_pim = flux_pr * omb;
        const float flux_psm = flux_pr * beta;
        const float flux_pby = flux_psm * oma;
        const float flux_pin = flux_psm * alpha;
        XIN += flux_pin;
        const float flux_ptf = fmaxf(XIN - stor, 0.0f);
        XIN = fmaxf(XIN - flux_ptf, NEARZERO);
        const float flux_ein = fminf(E * beta, XIN);
        XIN = fmaxf(XIN - flux_ein, NEARZERO);

        // ---- impervious store ----
        const float flux_mim = flux_m * omb;
        const float flux_msm = flux_m * beta;
        RSTOR += flux_mim + flux_pim;
        const float flux_sas = fmaxf(RSTOR - retip, 0.0f);
        RSTOR = fmaxf(RSTOR - flux_sas, NEARZERO);
        const float flux_eim = fminf(omb * E, RSTOR);
        RSTOR = fmaxf(RSTOR - flux_eim, NEARZERO);

        // ---- surface runoff / infiltration ----
        float ratio = fmaf(scx_m_scn, RECHR * inv_remx, scn);
        ratio = fminf(fmaxf(ratio, 0.0f), 1.0f);
        const float tosoil   = flux_msm + flux_ptf + flux_pby;
        const float flux_sro = ratio * tosoil;
        const float flux_inf = tosoil - flux_sro;

        // ---- recharge zone ----
        RECHR += flux_inf;
        const float flux_pc = fmaxf(RECHR - remx, 0.0f);
        RECHR -= flux_pc;
        const float ep_rem = E - flux_ein - flux_eim;
        const float evap_max_a = fmaxf(RECHR * inv_remx * ep_rem, 0.0f);
        const float flux_ea = fminf(evap_max_a, RECHR);
        RECHR = fmaxf(RECHR - flux_ea, NEARZERO);

        // ---- lower soil ----
        SMAV += flux_pc;
        const float flux_excs = fmaxf(SMAV - smax, 0.0f);
        SMAV -= flux_excs;
        float transp = (RECHR < ep_rem)
                         ? SMAV * inv_smax * (ep_rem - flux_ea) : 0.0f;
        transp = fmaxf(transp, 0.0f);
        SMAV = fmaxf(SMAV - transp, NEARZERO);

        // ---- reservoir ----
        const float flux_sep  = fminf(cgw, flux_excs);
        const float flux_qres = fmaxf(flux_excs - flux_sep, 0.0f);
        RES += flux_qres;
        // k1 * (RES/resmax)^k2 via TRANS unit: v_log_f32 (base-2) + v_exp_f32 (base-2)
        // RES >= NEARZERO > 0, so log argument is always positive.
        float flux_gad = k1 * __builtin_amdgcn_exp2f(
                                  k2 * __builtin_amdgcn_logf(RES * inv_resmax));
        flux_gad = fminf(flux_gad, RES);
        RES = fmaxf(RES - flux_gad, NEARZERO);
        float flux_ras = fmaf(k4 * RES, RES, k3 * RES);
        flux_ras = fminf(flux_ras, RES);
        RES = fmaxf(RES - flux_ras, NEARZERO);

        // ---- groundwater ----
        GW += flux_gad + flux_sep;
        const float flux_bas = k5 * GW;
        GW = fmaxf(GW - flux_bas, NEARZERO);
        const float flux_snk = k6 * GW;
        GW = fmaxf(GW - flux_snk, NEARZERO);

        const float Q = flux_sas + flux_sro + flux_bas + flux_ras;
        // Streaming output: non-temporal store hint
        __builtin_nontemporal_store(Q, &out[(size_t)t * total + tid]);

        P = Pn; T = Tn; E = En;
    }
}

extern "C" void kernel_launch(void* const* d_in, const int* in_sizes, int n_in,
                              void* d_out, int out_size, void* d_ws, size_t ws_size,
                              hipStream_t stream) {
    (void)n_in; (void)d_ws; (void)ws_size;
    const float* x      = (const float*)d_in[0];   // [Nstep, Ngrid, 3]
    const float* params = (const float*)d_in[1];   // [Nstep, Ngrid, 18, nmul]

    const int Nstep = 365;                                   // per reference
    const int Ngrid = in_sizes[0] / (3 * Nstep);             // 2000
    const int nmul  = in_sizes[1] / (Nstep * Ngrid * 18);    // 8

    // Only the last time-slice of params is used: params_raw[-1]
    const float* plast = params + (size_t)(Nstep - 1) * Ngrid * 18 * nmul;
    float* out = (float*)d_out;

    const int total   = Ngrid * nmul;
    const int block   = 256;                       // 8 wave32s per workgroup
    const int nblocks = (total + block - 1) / block;

    if (nmul == 8) {
        hipLaunchKernelGGL((prms_scan_kernel<8>), dim3(nblocks), dim3(block), 0, stream,
                           x, plast, out, Ngrid, nmul, Nstep);
    } else {
        hipLaunchKernelGGL((prms_scan_kernel<0>), dim3(nblocks), dim3(block), 0, stream,
                           x, plast, out, Ngrid, nmul, Nstep);
    }
}